// RGCN_27487790695081
// MI455X (gfx1250) — compile-verified
//
#include <hip/hip_runtime.h>

#define HD 128         // hidden dim (fixed by problem)
#define TM 16          // node tile (M) per block
#define MAXSLOT 5      // 4 bases + self-loop source

typedef float v2f __attribute__((ext_vector_type(2)));
typedef float v8f __attribute__((ext_vector_type(8)));

// ---------------------------------------------------------------------------
// Zero a float buffer (float4 stores).
// ---------------------------------------------------------------------------
__global__ void rgcn_zero_f4(float4* __restrict__ p, long long n4) {
    long long i = (long long)blockIdx.x * blockDim.x + threadIdx.x;
    if (i < n4) p[i] = make_float4(0.f, 0.f, 0.f, 0.f);
}

// ---------------------------------------------------------------------------
// Edge scatter: one wave32 per edge. Lane l owns float4 elements [4l..4l+3]
// of the 128-wide source feature. For each basis b in this pass:
//   s[b][dst] += (w_comp[etype,b0+b] * norm) * h[src]
// via hardware global_atomic_add_f32 (L2-resident accumulators).
// ---------------------------------------------------------------------------
__global__ __launch_bounds__(256)
void rgcn_edge_scatter(const float* __restrict__ h,
                       const float* __restrict__ norm,
                       const float* __restrict__ w_comp,
                       const int*   __restrict__ src,
                       const int*   __restrict__ dst,
                       const int*   __restrict__ etype,
                       float*       __restrict__ s,      // [nb][N][HD]
                       int E, int N, int b0, int nb)
{
    const int e = blockIdx.x * 8 + (threadIdx.x >> 5);
    if (e >= E) return;
    const int lane = threadIdx.x & 31;

    const int   sv = src[e];
    const int   dv = dst[e];
    const int   r  = etype[e];
    const float nm = norm[e];

    const float4 hv = ((const float4*)(h + (size_t)sv * HD))[lane];

    for (int j = 0; j < nb; ++j) {
        const float c = w_comp[r * 4 + b0 + j] * nm;
        float* p = s + ((size_t)j * N + dv) * HD + lane * 4;
        unsafeAtomicAdd(p + 0, hv.x * c);
        unsafeAtomicAdd(p + 1, hv.y * c);
        unsafeAtomicAdd(p + 2, hv.z * c);
        unsafeAtomicAdd(p + 3, hv.w * c);
    }
}

// ---------------------------------------------------------------------------
// Node GEMM with V_WMMA_F32_16X16X4_F32.
// Block = 256 threads = 8 waves. Block covers 16 node rows; wave w owns
// output columns [16w, 16w+16). A-sources (s_b tiles + h tile on last pass)
// are staged in LDS; B (weights) read from global (L2-resident, 320 KB total).
//
// Fragment layouts per CDNA5 ISA 7.12.2 (wave32):
//   A 16x4 f32 : a[v] = A[lane&15][(lane>>4)*2 + v]
//   B 4x16 f32 : b[v] = B[(lane>>4)*2 + v][lane&15]
//   C/D 16x16  : acc[i] -> row = (lane>>4)*8 + i, col = lane&15
// ---------------------------------------------------------------------------
__global__ __launch_bounds__(256)
void rgcn_node_gemm(const float* __restrict__ s,        // [nb][N][HD]
                    const float* __restrict__ h,        // [N][HD]
                    const float* __restrict__ basis_w,  // pre-offset: [nb][HD][HD]
                    const float* __restrict__ loop_w,   // [HD][HD]
                    const float* __restrict__ bias,     // [HD]
                    float*       __restrict__ out,      // [N][HD]
                    int N, int nb, int add_loop, int is_first)
{
    __shared__ float lds[MAXSLOT * TM * HD];            // 40 KB

    const int node0  = blockIdx.x * TM;
    const int t      = threadIdx.x;
    const int nslots = nb + (add_loop ? 1 : 0);

    // -- Stage A-sources: slots [0,nb) = s bases, slot nb = h (if add_loop) --
    const int f4_per_slot = TM * HD / 4;                // 512
    for (int idx = t; idx < nslots * f4_per_slot; idx += 256) {
        const int slot = idx / f4_per_slot;
        const int rem  = idx - slot * f4_per_slot;
        const int row  = rem >> 5;                      // 32 float4 per row
        const int c4   = rem & 31;
        const int node = node0 + row;
        float4 v = make_float4(0.f, 0.f, 0.f, 0.f);
        if (node < N) {
            const float* sp = (slot < nb)
                ? (s + ((size_t)slot * N + node) * HD)
                : (h + (size_t)node * HD);
            v = ((const float4*)sp)[c4];
        }
        ((float4*)&lds[(size_t)slot * (TM * HD) + row * HD])[c4] = v;
    }
    __syncthreads();

    const int lane  = t & 31;
    const int wave  = t >> 5;                           // n-tile 0..7
    const int col16 = lane & 15;
    const int khalf = lane >> 4;                        // 0 or 1
    const int ncol  = wave * 16 + col16;

    v8f acc = {};
    for (int m = 0; m < nslots; ++m) {
        const float* W = (m < nb) ? (basis_w + (size_t)m * HD * HD) : loop_w;
        const float* Abase = &lds[(size_t)m * (TM * HD) + col16 * HD + khalf * 2];
        #pragma unroll 4
        for (int k0 = 0; k0 < HD; k0 += 4) {
            v2f a;
            a.x = Abase[k0];
            a.y = Abase[k0 + 1];
            const int kk = k0 + khalf * 2;
            v2f b;
            b.x = W[(size_t)kk * HD + ncol];
            b.y = W[(size_t)(kk + 1) * HD + ncol];
            acc = __builtin_amdgcn_wmma_f32_16x16x4_f32(
                /*neg_a=*/false, a, /*neg_b=*/false, b,
                /*c_mod=*/(short)0, acc,
                /*reuse_a=*/false, /*reuse_b=*/false);
        }
    }

    // -- Epilogue: accumulate across passes; bias + ReLU on final pass --
    const float bv = bias[ncol];
    #pragma unroll
    for (int i = 0; i < 8; ++i) {
        const int node = node0 + khalf * 8 + i;
        if (node < N) {
            float* p = out + (size_t)node * HD + ncol;
            float v = acc[i];
            if (!is_first) v += *p;
            if (add_loop) { v += bv; v = v > 0.f ? v : 0.f; }
            *p = v;
        }
    }
}

// ---------------------------------------------------------------------------
// Host launcher. Inputs (setup_inputs order):
//  0 h[N,128] f32, 1 norm[E,1] f32, 2 basis_w[B,128,128] f32, 3 w_comp[R,B] f32,
//  4 loop_w[128,128] f32, 5 bias[128] f32, 6 src[E] i32, 7 dst[E] i32, 8 etype[E] i32
// Output: relu(agg + bias + h@loop_w)  [N,128] f32.
// ---------------------------------------------------------------------------
extern "C" void kernel_launch(void* const* d_in, const int* in_sizes, int n_in,
                              void* d_out, int out_size, void* d_ws, size_t ws_size,
                              hipStream_t stream)
{
    const float* h       = (const float*)d_in[0];
    const float* norm    = (const float*)d_in[1];
    const float* basis_w = (const float*)d_in[2];
    const float* w_comp  = (const float*)d_in[3];
    const float* loop_w  = (const float*)d_in[4];
    const float* bias    = (const float*)d_in[5];
    const int*   src     = (const int*)d_in[6];
    const int*   dst     = (const int*)d_in[7];
    const int*   etype   = (const int*)d_in[8];
    float*       out     = (float*)d_out;

    const int N  = in_sizes[0] / HD;                 // 50000
    const int E  = in_sizes[6];                      // 800000
    const int NB = in_sizes[2] / (HD * HD);          // 4 bases

    float* s = (float*)d_ws;

    // Deterministic pass sizing from fixed ws_size: how many per-basis
    // accumulators [N,128] f32 fit in workspace (clamped to [1, NB]).
    const size_t per_basis = (size_t)N * HD * sizeof(float);
    int bpp = (int)(ws_size / per_basis);
    if (bpp < 1)  bpp = 1;      // assume ws >= one accumulator (25.6 MB)
    if (bpp > NB) bpp = NB;

    const int node_blocks = (N + TM - 1) / TM;
    const int edge_blocks = (E + 7) / 8;

    for (int b0 = 0; b0 < NB; b0 += bpp) {
        const int nb       = (b0 + bpp <= NB) ? bpp : (NB - b0);
        const int is_first = (b0 == 0) ? 1 : 0;
        const int add_loop = (b0 + nb >= NB) ? 1 : 0;

        const long long n4 = (long long)nb * N * HD / 4;
        const int zblocks = (int)((n4 + 255) / 256);
        rgcn_zero_f4<<<zblocks, 256, 0, stream>>>((float4*)s, n4);

        rgcn_edge_scatter<<<edge_blocks, 256, 0, stream>>>(
            h, norm, w_comp, src, dst, etype, s, E, N, b0, nb);

        rgcn_node_gemm<<<node_blocks, 256, 0, stream>>>(
            s, h, basis_w + (size_t)b0 * HD * HD, loop_w, bias, out,
            N, nb, add_loop, is_first);
    }
}